// LSTMAutoencoder_36825049596434
// MI455X (gfx1250) — compile-verified
//
#include <hip/hip_runtime.h>
#include <math.h>

typedef __attribute__((ext_vector_type(2))) float v2f;
typedef __attribute__((ext_vector_type(8))) float v8f;

#define B_TOT   1024
#define T_LEN   512
#define ROWS    16      // batch rows per block
#define NTHREADS 512    // 16 wave32s
#define H_PAD   68      // pitch for h rows (4m bank spread, 16B aligned)
#define G_PAD   260     // pitch for gate rows (breaks 8-row bank aliasing)

__device__ __forceinline__ float sigm(float x) {
    return 1.0f / (1.0f + __expf(-x));
}
__device__ __forceinline__ float tanh_f(float x) {
    x = fminf(fmaxf(x, -15.0f), 15.0f);
    float e = __expf(2.0f * x);
    return (e - 1.0f) / (e + 1.0f);
}

__global__ __launch_bounds__(NTHREADS)
void lstm_ae_kernel(const float* __restrict__ x,
                    const float* __restrict__ eWih, const float* __restrict__ eWhh,
                    const float* __restrict__ ebih, const float* __restrict__ ebhh,
                    const float* __restrict__ dWih, const float* __restrict__ dWhh,
                    const float* __restrict__ dbih, const float* __restrict__ dbhh,
                    float* __restrict__ out)
{
    __shared__ __align__(16) float ldsG[ROWS * G_PAD];   // encoder gates 16x256 (padded)
    __shared__ __align__(16) float ldsH[ROWS * H_PAD];   // h_enc 16x64 (padded)
    __shared__ __align__(16) float ldsDecW[16 * H_PAD];  // dec_Wih 12x64 (padded to 16 rows)
    __shared__ __align__(16) float ldsX[2 * ROWS * 4];   // x tile double buffer 16x3 (pad 4)
    __shared__ __align__(16) float ldsHdec[ROWS * 4];    // h_dec 16x3 (pad 4)

    const int tid  = threadIdx.x;
    const int lane = tid & 31;
    const int w    = tid >> 5;          // wave id 0..15 == encoder col-tile == gate row owner
    const int hi   = lane >> 4;         // lane half (K split for f32 WMMA)
    const int l16  = lane & 15;
    const int b0   = blockIdx.x * ROWS;

    // ---- encoder per-lane loop-invariant constants ----
    const int ncol = w * 16 + l16;      // gate column 0..255 this lane owns in C
    const float wih0 = eWih[ncol * 3 + 0];
    const float wih1 = eWih[ncol * 3 + 1];
    const float wih2 = eWih[ncol * 3 + 2];
    const float bsum = ebih[ncol] + ebhh[ncol];

    // B fragments for V_WMMA_F32_16X16X4_F32: lane holds B[k0+2*hi .. +1][n=l16]
    // == Whh[ncol][k0+2*hi], Whh[ncol][k0+2*hi+1]; K=64 -> 16 fragments.
    v2f bfrag[16];
#pragma unroll
    for (int kk = 0; kk < 16; ++kk) {
        const int k = 4 * kk + 2 * hi;
        bfrag[kk].x = eWhh[ncol * 64 + k];
        bfrag[kk].y = eWhh[ncol * 64 + k + 1];
    }

    // ---- decoder per-thread constants (threads < 256: row dm, gate col dn) ----
    const int dm = tid >> 4;
    const int dn = tid & 15;            // valid gates are dn < 12 (i,f,g,o x 3)
    float dwhh0 = 0.f, dwhh1 = 0.f, dwhh2 = 0.f, dbsum = 0.f;
    if (tid < 256) {
        const int nn = (dn < 12) ? dn : 0;
        dwhh0 = dWhh[nn * 3 + 0];
        dwhh1 = dWhh[nn * 3 + 1];
        dwhh2 = dWhh[nn * 3 + 2];
        dbsum = dbih[nn] + dbhh[nn];
    }

    // ---- init LDS ----
    for (int i = tid; i < ROWS * H_PAD; i += NTHREADS) ldsH[i] = 0.0f;
    for (int i = tid; i < 16 * H_PAD; i += NTHREADS) {
        const int n = i / H_PAD, k = i - n * H_PAD;
        ldsDecW[i] = (n < 12 && k < 64) ? dWih[n * 64 + k] : 0.0f;
    }
    if (tid < ROWS * 4) ldsHdec[tid] = 0.0f;
    if (tid < 48) {                      // x tile for t = 0
        const int m = tid / 3, d = tid - 3 * m;
        ldsX[m * 4 + d] = x[(b0 + m) * (T_LEN * 3) + d];
    }
    float ce0 = 0.0f, ce1 = 0.0f;        // encoder cell state (2 hidden elems/thread)
    float cd  = 0.0f;                    // decoder cell state (threads<256, dn<3)
    __syncthreads();

    for (int t = 0; t < T_LEN; ++t) {
        const int cur = t & 1;

        // -------- Phase A: gates = bias + x(t)@Wih^T + h(t-1)@Whh^T via WMMA --------
        v8f acc;
        {
            const float* xs = &ldsX[cur * 64];
#pragma unroll
            for (int r = 0; r < 8; ++r) {
                const int m = r + 8 * hi;    // C layout: vgpr r -> M=r (lo lanes) / r+8 (hi lanes)
                acc[r] = bsum + xs[m * 4 + 0] * wih0
                              + xs[m * 4 + 1] * wih1
                              + xs[m * 4 + 2] * wih2;
            }
        }
        {
            // A fragment: lane holds h[M=l16][k0+2*hi], h[M=l16][k0+2*hi+1]
            const float* hA = &ldsH[l16 * H_PAD + 2 * hi];
#pragma unroll
            for (int kk = 0; kk < 16; ++kk) {
                v2f a;
                a.x = hA[4 * kk];
                a.y = hA[4 * kk + 1];
                acc = __builtin_amdgcn_wmma_f32_16x16x4_f32(
                        false, a, false, bfrag[kk], (short)0, acc, false, false);
            }
        }
        {
            float* g = &ldsG[w * 16 + l16];
#pragma unroll
            for (int r = 0; r < 8; ++r) {
                const int m = r + 8 * hi;
                g[m * G_PAD] = acc[r];
            }
        }
        if (tid < 48 && t + 1 < T_LEN) {   // prefetch next x tile into other buffer
            const int m = tid / 3, d = tid - 3 * (tid / 3);
            ldsX[(1 - cur) * 64 + m * 4 + d] =
                x[(b0 + m) * (T_LEN * 3) + (t + 1) * 3 + d];
        }
        __syncthreads();

        // -------- Phase B: encoder cell/hidden update (wave w owns gate row m=w) --------
        {
            const float* row = &ldsG[w * G_PAD];
            const int j = lane * 2;
            const float i0 = sigm(row[j]),         i1 = sigm(row[j + 1]);
            const float f0 = sigm(row[64 + j]),    f1 = sigm(row[64 + j + 1]);
            const float g0 = tanh_f(row[128 + j]), g1 = tanh_f(row[128 + j + 1]);
            const float o0 = sigm(row[192 + j]),   o1 = sigm(row[192 + j + 1]);
            ce0 = f0 * ce0 + i0 * g0;
            ce1 = f1 * ce1 + i1 * g1;
            ldsH[w * H_PAD + j]     = o0 * tanh_f(ce0);
            ldsH[w * H_PAD + j + 1] = o1 * tanh_f(ce1);
        }
        __syncthreads();

        // -------- Phase C: fused decoder step (waves 0..7; 16 lanes per batch row) ----
        if (tid < 256) {
            float a = dbsum + ldsHdec[dm * 4 + 0] * dwhh0
                            + ldsHdec[dm * 4 + 1] * dwhh1
                            + ldsHdec[dm * 4 + 2] * dwhh2;
            const float4* hr = (const float4*)&ldsH[dm * H_PAD];
            const float4* wr = (const float4*)&ldsDecW[dn * H_PAD];
#pragma unroll
            for (int q = 0; q < 16; ++q) {
                const float4 hv = hr[q], wv = wr[q];
                a += hv.x * wv.x + hv.y * wv.y + hv.z * wv.z + hv.w * wv.w;
            }
            // gather i,f,g,o for hidden j from lanes base+{j,3+j,6+j,9+j} of this row
            const int base = lane & 16;
            const int j3   = dn % 3;
            const float gi = __shfl(a, base + j3,     32);
            const float gf = __shfl(a, base + 3 + j3, 32);
            const float gg = __shfl(a, base + 6 + j3, 32);
            const float go = __shfl(a, base + 9 + j3, 32);
            if (dn < 3) {
                const float iv = sigm(gi), fv = sigm(gf);
                const float gv = tanh_f(gg), ov = sigm(go);
                cd = fv * cd + iv * gv;
                const float hd = ov * tanh_f(cd);
                ldsHdec[dm * 4 + dn] = hd;
                out[(b0 + dm) * (T_LEN * 3) + t * 3 + dn] = hd;
            }
        }
    }
}

extern "C" void kernel_launch(void* const* d_in, const int* in_sizes, int n_in,
                              void* d_out, int out_size, void* d_ws, size_t ws_size,
                              hipStream_t stream) {
    const float* x    = (const float*)d_in[0];
    const float* eWih = (const float*)d_in[1];
    const float* eWhh = (const float*)d_in[2];
    const float* ebih = (const float*)d_in[3];
    const float* ebhh = (const float*)d_in[4];
    const float* dWih = (const float*)d_in[5];
    const float* dWhh = (const float*)d_in[6];
    const float* dbih = (const float*)d_in[7];
    const float* dbhh = (const float*)d_in[8];
    float* out = (float*)d_out;

    dim3 grid(B_TOT / ROWS);     // 64 blocks, one 16-row batch tile each
    dim3 block(NTHREADS);        // 16 wave32s
    hipLaunchKernelGGL(lstm_ae_kernel, grid, block, 0, stream,
                       x, eWih, eWhh, ebih, ebhh, dWih, dWhh, dbih, dbhh, out);
}